// RNNModel_1709396984259
// MI455X (gfx1250) — compile-verified
//
#include <hip/hip_runtime.h>
#include <hip/hip_bf16.h>

// CDNA5 (gfx1250) wave32 LSTM.
//   gates^T = [W_hh | W_ih | bias] x [h^T ; x^T ; 1] via chained
//   v_wmma_f32_16x16x4_f32 (M=16 gate rows, N=16 batch cols, K=4 chunks).
// Gate rows of the A operand are PERMUTED (swap bits 2<->3 of the row index:
// [i0..3, f0..3, i4..7, f4..7] and [g0..3, o0..3, g4..7, o4..7]) so the WMMA
// output lands PRE-PACKED: acc0[k]=i, acc0[k+4]=f, acc1[k]=g, acc1[k+4]=o with
// k=0..3 in lanes 0-15 and k=4..7 in lanes 16-31. State h/c is register-
// resident in the same packed layout -> activations use all 32 lanes and gate
// extraction needs ZERO cross-lane moves; only the B-operand rebuild needs
// 4 ds_swizzle xor-16 per step.

typedef __attribute__((ext_vector_type(2))) float v2f;
typedef __attribute__((ext_vector_type(8))) float v8f;

#define SWZ_XOR16 0x401f  // group-of-32 swizzle: and=0x1f, or=0, xor=0x10

__device__ __forceinline__ float swz16(float v) {
  return __int_as_float(__builtin_amdgcn_ds_swizzle(__float_as_int(v), SWZ_XOR16));
}
__device__ __forceinline__ float sigm(float x) {
  // rcp(1 + 2^(-log2e*x)) : mul, exp, add, rcp
  float e = __builtin_amdgcn_exp2f(x * -1.4426950408889634f);
  return __builtin_amdgcn_rcpf(1.0f + e);
}
__device__ __forceinline__ float ftanh(float x) {
  // fma(2, rcp(1 + 2^(-2*log2e*x)), -1) : mul, exp, add, rcp, fma
  float e = __builtin_amdgcn_exp2f(x * -2.8853900817779268f);
  return __builtin_fmaf(2.0f, __builtin_amdgcn_rcpf(1.0f + e), -1.0f);
}
// Row permutation: swap bit2 <-> bit3 (maps rows [0..15] -> gates
// [0..3, 8..11, 4..7, 12..15] so i/f (g/o) come out half-wave packed).
__device__ __forceinline__ int permgate(int j) {
  return (j & 3) | ((j & 4) << 1) | ((j & 8) >> 1);
}

__global__ __launch_bounds__(32) void lstm_wmma_kernel(
    const float* __restrict__ x, const float* __restrict__ W_ih,
    const float* __restrict__ W_hh, const float* __restrict__ b_ih,
    const float* __restrict__ b_hh, const float* __restrict__ W_fc,
    const float* __restrict__ b_fc, float* __restrict__ out, int T) {
  const int lane = threadIdx.x & 31;
  const bool lo = lane < 16;
  const int j0 = lane & 15;   // A: M-row index / B,D: N-col (batch) index
  const int ko = lo ? 0 : 2;  // f32 A/B layout: high half-wave holds K+2
  const int tile = blockIdx.x;

  // ---- Time-invariant A operands (permuted gate rows) ----
  const int pg0 = permgate(j0);       // D0 rows: i|f permuted, gates 0..15
  const int pg1 = permgate(j0) + 16;  // D1 rows: g|o permuted, gates 16..31
  v2f A00, A01, A10, A11, Ax0, Ax1;
  A00.x = W_hh[pg0 * 8 + ko + 0];  // K=0..3 chunk
  A00.y = W_hh[pg0 * 8 + ko + 1];
  A01.x = W_hh[pg0 * 8 + ko + 4];  // K=4..7 chunk
  A01.y = W_hh[pg0 * 8 + ko + 5];
  A10.x = W_hh[pg1 * 8 + ko + 0];
  A10.y = W_hh[pg1 * 8 + ko + 1];
  A11.x = W_hh[pg1 * 8 + ko + 4];
  A11.y = W_hh[pg1 * 8 + ko + 5];
  // Rank-2 augmented chunk folding input weight + (b_ih+b_hh).
  Ax0.x = lo ? W_ih[pg0] : 0.0f;
  Ax0.y = lo ? (b_ih[pg0] + b_hh[pg0]) : 0.0f;
  Ax1.x = lo ? W_ih[pg1] : 0.0f;
  Ax1.y = lo ? (b_ih[pg1] + b_hh[pg1]) : 0.0f;

  const float one_lo = lo ? 1.0f : 0.0f;  // B'' K=1 row (bias path)

  const int row = tile * 16 + j0;  // batch row this lane streams / stores
  const float* xrow = x + (size_t)row * (size_t)T;

  // Packed state: VGPR k: lane m -> h[m,k], lane m+16 -> h[m,k+4].
  float hs[4], cs[4];
#pragma unroll
  for (int k = 0; k < 4; ++k) { hs[k] = 0.0f; cs[k] = 0.0f; }

  for (int t0 = 0; t0 < T; t0 += 4) {
    const float4 xq = *(const float4*)(xrow + t0);  // contiguous-in-t stream
#pragma unroll
    for (int u = 0; u < 4; ++u) {
      const float xv = (u == 0) ? xq.x : (u == 1) ? xq.y : (u == 2) ? xq.z : xq.w;

      // ---- B operand: h^T in WMMA B layout (4 swizzles + 4 selects) ----
      v2f Blo, Bhi, Bx;
      Blo.x = lo ? hs[0] : swz16(hs[2]);  // K=0 | K=2
      Blo.y = lo ? hs[1] : swz16(hs[3]);  // K=1 | K=3
      Bhi.x = lo ? swz16(hs[0]) : hs[2];  // K=4 | K=6
      Bhi.y = lo ? swz16(hs[1]) : hs[3];  // K=5 | K=7
      Bx.x = lo ? xv : 0.0f;              // B'' = [x^T ; 1 ; 0 ; 0]
      Bx.y = one_lo;

      // ---- gates^T : 3 chained WMMAs per gate half ----
      v8f z = {};
      v8f acc0 = __builtin_amdgcn_wmma_f32_16x16x4_f32(false, Ax0, false, Bx, (short)0, z, false, false);
      acc0 = __builtin_amdgcn_wmma_f32_16x16x4_f32(false, A00, false, Blo, (short)0, acc0, false, false);
      acc0 = __builtin_amdgcn_wmma_f32_16x16x4_f32(false, A01, false, Bhi, (short)0, acc0, false, false);
      v8f acc1 = __builtin_amdgcn_wmma_f32_16x16x4_f32(false, Ax1, false, Bx, (short)0, z, false, false);
      acc1 = __builtin_amdgcn_wmma_f32_16x16x4_f32(false, A10, false, Blo, (short)0, acc1, false, false);
      acc1 = __builtin_amdgcn_wmma_f32_16x16x4_f32(false, A11, false, Bhi, (short)0, acc1, false, false);

      // ---- LSTM update: gates arrive pre-packed, all 32 lanes useful ----
#pragma unroll
      for (int k = 0; k < 4; ++k) {
        const float iv = sigm(acc0[k]);
        const float fv = sigm(acc0[k + 4]);
        const float gv = ftanh(acc1[k]);
        const float ov = sigm(acc1[k + 4]);
        cs[k] = __builtin_fmaf(fv, cs[k], iv * gv);
        hs[k] = ov * ftanh(cs[k]);
      }
    }
  }

  // ---- Final projection: out[m, j] = b_fc[j] + sum_k h[m,k] * W_fc[j,k] ----
  // Each half-wave reduces its 4 k's, then one xor-16 swizzle combines halves.
  const int kofs = lo ? 0 : 4;
#pragma unroll
  for (int j = 0; j < 3; ++j) {
    float pj = 0.0f;
#pragma unroll
    for (int kk = 0; kk < 4; ++kk)
      pj = __builtin_fmaf(hs[kk], W_fc[j * 8 + kk + kofs], pj);
    pj += swz16(pj);  // full-EXEC swizzle: combine half-wave partial sums
    if (lo) out[row * 3 + j] = pj + b_fc[j];
  }
}

extern "C" void kernel_launch(void* const* d_in, const int* in_sizes, int n_in,
                              void* d_out, int out_size, void* d_ws, size_t ws_size,
                              hipStream_t stream) {
  const float* x    = (const float*)d_in[0];
  const float* W_ih = (const float*)d_in[1];
  const float* W_hh = (const float*)d_in[2];
  const float* b_ih = (const float*)d_in[3];
  const float* b_hh = (const float*)d_in[4];
  const float* W_fc = (const float*)d_in[5];
  const float* b_fc = (const float*)d_in[6];
  float* out = (float*)d_out;

  const int T = 512;
  const int B = in_sizes[0] / T;    // 8192
  const int tiles = (B + 15) / 16;  // 512 waves, one 16-batch tile each

  lstm_wmma_kernel<<<dim3(tiles), dim3(32), 0, stream>>>(
      x, W_ih, W_hh, b_ih, b_hh, W_fc, b_fc, out, T);
}